// LuongDecoderRNN_81372450390338
// MI455X (gfx1250) — compile-verified
//
#include <hip/hip_runtime.h>
#include <math.h>

typedef __attribute__((ext_vector_type(2))) float v2f;
typedef __attribute__((ext_vector_type(8))) float v8f;

#define HH   1024
#define VV   50257
#define BB   64
#define TT   2048
#define VPAD 50272
#define KCH  128
#define LDAP (KCH + 4)   /* 132 floats: 32-lane b64 frag reads cover all 64 banks */
#define NSL  32          /* attention t-slices */
#define TSL  (TT / NSL)  /* 64 timesteps per slice */

// CDNA5 async bulk copy: global -> LDS, 16B per lane, tracked by ASYNCcnt.
// VDST = LDS byte address (low 32 bits of generic pointer), VADDR = global addr.
__device__ __forceinline__ void async_copy16(const float* __restrict__ g, float* l)
{
  unsigned lo = (unsigned)(uintptr_t)l;
  asm volatile("global_load_async_to_lds_b128 %0, %1, off" :: "v"(lo), "v"(g) : "memory");
}
__device__ __forceinline__ void async_wait0()
{
  asm volatile("s_wait_asynccnt 0x0" ::: "memory");
}

// ---------------------------------------------------------------------------
// C[64,N] = act( A[64,K] @ W[N,K]^T + bias ), fp32 WMMA 16x16x4
// block = 128 threads (4 waves). blockIdx.x = N-tile. wave = M-tile.
// ---------------------------------------------------------------------------
__global__ __launch_bounds__(128) void gemm64_nt(
    const float* __restrict__ A, const float* __restrict__ W,
    const float* __restrict__ bias, float* __restrict__ C,
    int K, int Nrows, int ldc, int act)
{
  __shared__ float Al[64 * LDAP];
  __shared__ float Wl[16 * LDAP];
  const int tid  = threadIdx.x;
  const int lane = tid & 31;
  const int wave = tid >> 5;        // M-tile 0..3
  const int half = lane >> 4;       // 0 / 1
  const int r16  = lane & 15;
  const int n0   = blockIdx.x * 16;

  v8f acc = {};
  const int nch = K / KCH;
  for (int c = 0; c < nch; ++c) {
    const int kb = c * KCH;
    // stage A chunk: 64 x 128 floats, async 16B copies (2048 f4, 16/thread)
    for (int i = tid; i < 64 * (KCH / 4); i += 128) {
      const int row = i >> 5;
      const int cv  = (i & 31) << 2;
      async_copy16(A + (size_t)row * K + kb + cv, &Al[row * LDAP + cv]);
    }
    // stage W chunk: 16 x 128 floats (row-clamped for the Wo tail tile)
    for (int i = tid; i < 16 * (KCH / 4); i += 128) {
      const int row = i >> 5;
      const int cv  = (i & 31) << 2;
      int gr = n0 + row; if (gr >= Nrows) gr = Nrows - 1;
      async_copy16(W + (size_t)gr * K + kb + cv, &Wl[row * LDAP + cv]);
    }
    // prefetch next W chunk (global_prefetch_b8)
    if (c + 1 < nch && tid < 16) {
      int gr = n0 + tid; if (gr >= Nrows) gr = Nrows - 1;
      __builtin_prefetch(W + (size_t)gr * K + kb + KCH, 0, 1);
    }
    async_wait0();
    __syncthreads();

    const float* ap = &Al[(wave * 16 + r16) * LDAP + 2 * half];
    const float* bp = &Wl[r16 * LDAP + 2 * half];
#pragma unroll
    for (int k = 0; k < KCH; k += 4) {
      v2f a = *(const v2f*)(ap + k);   // A 16x4: VGPR0=K{0,2}, VGPR1=K{1,3}
      v2f b = *(const v2f*)(bp + k);   // B 4x16: same split at lane 16
      acc = __builtin_amdgcn_wmma_f32_16x16x4_f32(
          false, a, false, b, (short)0, acc, false, false);
    }
    __syncthreads();
  }

  const int col = n0 + r16;
  if (col < Nrows) {
    const float bv = bias ? bias[col] : 0.0f;
#pragma unroll
    for (int r = 0; r < 8; ++r) {
      const int orow = wave * 16 + r + 8 * half;  // D: VGPR r -> M = r + 8*half
      float v = acc[r] + bv;
      if (act) v = tanhf(v);
      C[(size_t)orow * ldc + col] = v;
    }
  }
}

// ---------------------------------------------------------------------------
__global__ void embed_concat(const int* __restrict__ inputs,
                             const float* __restrict__ E,
                             const float* __restrict__ context,
                             float* __restrict__ x)
{
  const int idx = blockIdx.x * blockDim.x + threadIdx.x;  // 64*2048
  const int b = idx >> 11, c = idx & 2047;
  x[idx] = (c < HH) ? E[(size_t)inputs[b] * HH + c]
                    : context[(size_t)b * HH + (c - HH)];
}

__global__ void gru_gates(const float* __restrict__ gi, const float* __restrict__ gh,
                          const float* __restrict__ hp, float* __restrict__ h)
{
  const int idx = blockIdx.x * blockDim.x + threadIdx.x;  // 64*1024
  const int b = idx >> 10, j = idx & 1023;
  const float* gib = gi + (size_t)b * 3072;
  const float* ghb = gh + (size_t)b * 3072;
  const float r = 1.f / (1.f + expf(-(gib[j] + ghb[j])));
  const float z = 1.f / (1.f + expf(-(gib[HH + j] + ghb[HH + j])));
  const float n = tanhf(gib[2 * HH + j] + r * ghb[2 * HH + j]);
  h[idx] = (1.f - z) * n + z * hp[idx];
}

__global__ void resid_hidden(const float* __restrict__ h0, const float* __restrict__ h1,
                             const float* __restrict__ x, float* __restrict__ nh)
{
  const int idx = blockIdx.x * blockDim.x + threadIdx.x;  // 2*64*1024
  const int l = idx >> 16, rem = idx & 65535;
  const int b = rem >> 10, hh = rem & 1023;
  nh[idx] = (l ? h1 : h0)[rem] + x[(size_t)b * 2048 + hh];  // embedded = x[:, :H]
}

__global__ void transpose_sq(const float* __restrict__ src, float* __restrict__ dst, int n)
{
  const int idx = blockIdx.x * blockDim.x + threadIdx.x;
  const int i = idx / n, j = idx % n;
  dst[(size_t)j * n + i] = src[idx];
}

__global__ __launch_bounds__(256) void dot_ba(const float* __restrict__ h1,
                                              const float* __restrict__ ba,
                                              float* __restrict__ cvec)
{
  __shared__ float red[256];
  const int b = blockIdx.x, tid = threadIdx.x;
  float s = 0.f;
  for (int h = tid; h < HH; h += 256) s += h1[(size_t)b * HH + h] * ba[h];
  red[tid] = s; __syncthreads();
  for (int o = 128; o > 0; o >>= 1) { if (tid < o) red[tid] += red[tid + o]; __syncthreads(); }
  if (tid == 0) cvec[b] = red[0];
}

// ---------------------------------------------------------------------------
// One-pass attention: wave = (b, slice). Each enc row is read ONCE and used
// for both the score dot-product and the online-softmax context accumulation.
// Emits raw scores (for exact attn + global M,S) and per-slice (ctx, max).
// ---------------------------------------------------------------------------
__global__ __launch_bounds__(256) void attn_flash(
    const float* __restrict__ u, const float* __restrict__ enc,
    const float* __restrict__ cvec,
    float* __restrict__ scores,   // [B,T]
    float* __restrict__ part,     // [NSL,B,H] unnormalized local ctx
    float* __restrict__ pm)       // [NSL,B]   local max
{
  const int w    = blockIdx.x * 8 + (threadIdx.x >> 5);  // 0 .. B*NSL-1
  const int lane = threadIdx.x & 31;
  const int b    = w >> 5;        // / NSL
  const int s    = w & (NSL - 1);
  const float cb = cvec[b];

  const float4* uu = (const float4*)(u + (size_t)b * HH);
  float4 ur[8];
#pragma unroll
  for (int i = 0; i < 8; ++i) ur[i] = uu[lane + 32 * i];

  float m = -INFINITY;
  float4 acc[8];
#pragma unroll
  for (int i = 0; i < 8; ++i) acc[i] = make_float4(0.f, 0.f, 0.f, 0.f);

  const int t0 = s * TSL;
  for (int t = t0; t < t0 + TSL; ++t) {
    const float4* e = (const float4*)(enc + ((size_t)t * BB + b) * HH);
    float4 ev[8];
#pragma unroll
    for (int i = 0; i < 8; ++i) ev[i] = e[lane + 32 * i];
    float d = 0.f;
#pragma unroll
    for (int i = 0; i < 8; ++i)
      d += ev[i].x * ur[i].x + ev[i].y * ur[i].y + ev[i].z * ur[i].z + ev[i].w * ur[i].w;
#pragma unroll
    for (int o = 16; o > 0; o >>= 1) d += __shfl_xor(d, o, 32);  // all lanes get score
    d += cb;
    if (lane == 0) scores[(size_t)b * TT + t] = d;

    const float nm = fmaxf(m, d);
    const float f  = expf(m - nm);   // 0 on first step (m = -inf)
    const float e1 = expf(d - nm);
#pragma unroll
    for (int i = 0; i < 8; ++i) {
      acc[i].x = acc[i].x * f + e1 * ev[i].x;
      acc[i].y = acc[i].y * f + e1 * ev[i].y;
      acc[i].z = acc[i].z * f + e1 * ev[i].z;
      acc[i].w = acc[i].w * f + e1 * ev[i].w;
    }
    m = nm;
  }

  float4* p = (float4*)(part + ((size_t)s * BB + b) * HH);
#pragma unroll
  for (int i = 0; i < 8; ++i) p[lane + 32 * i] = acc[i];
  if (lane == 0) pm[s * BB + b] = m;
}

// softmax over T; also exports global max M and sum S per row for ctx combine
__global__ __launch_bounds__(256) void softmax_T(const float* __restrict__ scores,
                                                 float* __restrict__ attn,
                                                 float* __restrict__ MS)  // [2*B]
{
  __shared__ float red[256];
  const int b = blockIdx.x, tid = threadIdx.x;
  const float* s = scores + (size_t)b * TT;
  float m = -INFINITY;
  for (int t = tid; t < TT; t += 256) m = fmaxf(m, s[t]);
  red[tid] = m; __syncthreads();
  for (int o = 128; o > 0; o >>= 1) { if (tid < o) red[tid] = fmaxf(red[tid], red[tid + o]); __syncthreads(); }
  m = red[0]; __syncthreads();
  float sum = 0.f;
  for (int t = tid; t < TT; t += 256) { const float e = expf(s[t] - m); attn[(size_t)b * TT + t] = e; sum += e; }
  red[tid] = sum; __syncthreads();
  for (int o = 128; o > 0; o >>= 1) { if (tid < o) red[tid] += red[tid + o]; __syncthreads(); }
  const float S = red[0];
  const float inv = 1.f / S;
  for (int t = tid; t < TT; t += 256) attn[(size_t)b * TT + t] *= inv;
  if (tid == 0) { MS[b] = m; MS[BB + b] = S; }
}

// ctx[b,h] = (sum_s part[s,b,h] * exp(pm[s,b] - M[b])) / S[b]; also builds xo
__global__ void ctx_flash_reduce(const float* __restrict__ part, const float* __restrict__ pm,
                                 const float* __restrict__ MS, const float* __restrict__ h1,
                                 float* __restrict__ ctx_out, float* __restrict__ xo)
{
  const int idx = blockIdx.x * blockDim.x + threadIdx.x;  // 64*1024
  const int b = idx >> 10, h = idx & 1023;
  const float M = MS[b];
  const float invS = 1.f / MS[BB + b];
  float ssum = 0.f;
#pragma unroll
  for (int s = 0; s < NSL; ++s)
    ssum += part[((size_t)s * BB + b) * HH + h] * expf(pm[s * BB + b] - M);
  const float v = ssum * invS;
  ctx_out[idx] = v;
  xo[(size_t)b * 2048 + HH + h] = v;
  xo[(size_t)b * 2048 + h] = h1[idx];
}

__global__ __launch_bounds__(512) void logsoftmax_V(const float* __restrict__ logits,
                                                    float* __restrict__ out)
{
  __shared__ float red[512];
  const int b = blockIdx.x, tid = threadIdx.x;
  const float* l = logits + (size_t)b * VPAD;
  float m = -INFINITY;
  for (int v = tid; v < VV; v += 512) m = fmaxf(m, l[v]);
  red[tid] = m; __syncthreads();
  for (int o = 256; o > 0; o >>= 1) { if (tid < o) red[tid] = fmaxf(red[tid], red[tid + o]); __syncthreads(); }
  m = red[0]; __syncthreads();
  float sum = 0.f;
  for (int v = tid; v < VV; v += 512) sum += expf(l[v] - m);
  red[tid] = sum; __syncthreads();
  for (int o = 256; o > 0; o >>= 1) { if (tid < o) red[tid] += red[tid + o]; __syncthreads(); }
  const float lse = m + logf(red[0]);
  for (int v = tid; v < VV; v += 512) out[(size_t)b * VV + v] = l[v] - lse;
}

// ---------------------------------------------------------------------------
extern "C" void kernel_launch(void* const* d_in, const int* in_sizes, int n_in,
                              void* d_out, int out_size, void* d_ws, size_t ws_size,
                              hipStream_t stream)
{
  const int*   inputs  = (const int*)d_in[0];
  const float* hidden  = (const float*)d_in[1];   // [2,64,1024]
  const float* context = (const float*)d_in[2];   // [64,1024]
  const float* enc     = (const float*)d_in[3];   // [2048,64,1024]
  const float* E       = (const float*)d_in[4];   // [V,1024]
  const float* W_ih0   = (const float*)d_in[5];   // [3072,2048]
  const float* W_hh0   = (const float*)d_in[6];   // [3072,1024]
  const float* b_ih0   = (const float*)d_in[7];
  const float* b_hh0   = (const float*)d_in[8];
  const float* W_ih1   = (const float*)d_in[9];   // [3072,1024]
  const float* W_hh1   = (const float*)d_in[10];  // [3072,1024]
  const float* b_ih1   = (const float*)d_in[11];
  const float* b_hh1   = (const float*)d_in[12];
  const float* Wa      = (const float*)d_in[13];  // [1024,1024]
  const float* ba      = (const float*)d_in[14];
  const float* Wo      = (const float*)d_in[15];  // [V,2048]
  const float* bo      = (const float*)d_in[16];

  float* ws = (float*)d_ws;
  float* x      = ws;                 // 64*2048
  float* gi     = x      + 131072;    // 64*3072
  float* gh     = gi     + 196608;    // 64*3072
  float* h0     = gh     + 196608;    // 64*1024
  float* h1     = h0     + 65536;     // 64*1024
  float* WaT    = h1     + 65536;     // 1024*1024
  float* u      = WaT    + 1048576;   // 64*1024
  float* cvec   = u      + 65536;     // 64
  float* scores = cvec   + 64;        // 64*2048
  float* pm     = scores + 131072;    // 32*64
  float* MS     = pm     + 2048;      // 2*64
  float* part   = MS     + 128;       // 32*64*1024
  float* xo     = part   + 2097152;   // 64*2048
  float* logits = xo     + 131072;    // 64*50272

  float* out_lsm = (float*)d_out;                 // [64, V]
  float* out_nh  = out_lsm + (size_t)BB * VV;     // [2,64,1024]
  float* out_ctx = out_nh  + 2 * BB * HH;         // [64,1024]
  float* out_att = out_ctx + BB * HH;             // [64,1,2048]

  // x = [embed | context]
  embed_concat<<<512, 256, 0, stream>>>(inputs, E, context, x);

  // GRU layer 0
  gemm64_nt<<<192, 128, 0, stream>>>(x, W_ih0, b_ih0, gi, 2048, 3072, 3072, 0);
  gemm64_nt<<<192, 128, 0, stream>>>(hidden, W_hh0, b_hh0, gh, 1024, 3072, 3072, 0);
  gru_gates<<<256, 256, 0, stream>>>(gi, gh, hidden, h0);

  // GRU layer 1
  gemm64_nt<<<192, 128, 0, stream>>>(h0, W_ih1, b_ih1, gi, 1024, 3072, 3072, 0);
  gemm64_nt<<<192, 128, 0, stream>>>(hidden + (size_t)BB * HH, W_hh1, b_hh1, gh, 1024, 3072, 3072, 0);
  gru_gates<<<256, 256, 0, stream>>>(gi, gh, hidden + (size_t)BB * HH, h1);

  // new_hidden = stack(h0,h1) + embedded
  resid_hidden<<<512, 256, 0, stream>>>(h0, h1, x, out_nh);

  // attention precompute: u = h1 @ Wa (via WaT), c = h1 . ba
  transpose_sq<<<4096, 256, 0, stream>>>(Wa, WaT, 1024);
  gemm64_nt<<<64, 128, 0, stream>>>(h1, WaT, nullptr, u, 1024, 1024, 1024, 0);
  dot_ba<<<64, 256, 0, stream>>>(h1, ba, cvec);

  // one-pass attention: enc read once for scores AND context partials
  attn_flash<<<256, 256, 0, stream>>>(u, enc, cvec, scores, part, pm);
  softmax_T<<<64, 256, 0, stream>>>(scores, out_att, MS);
  ctx_flash_reduce<<<256, 256, 0, stream>>>(part, pm, MS, h1, out_ctx, xo);

  // logits = tanh(xo @ Wo^T + bo), then log_softmax over V
  gemm64_nt<<<(VV + 15) / 16, 128, 0, stream>>>(xo, Wo, bo, logits, 2048, VV, VPAD, 1);
  logsoftmax_V<<<64, 512, 0, stream>>>(logits, out_lsm);
}